// AdaptiveLinearWithChannel_11982958756441
// MI455X (gfx1250) — compile-verified
//
#include <hip/hip_runtime.h>

// Problem constants (match reference)
#define T_TILES 512
#define P_PTS   512
#define D_IN    256
#define D_OUT   256

// K-slab staged in LDS per iteration
#define KB        32
#define XS_STRIDE (KB + 4)           // 36 floats; conflict-free strided reads
#define WT_STRIDE (KB + 4)           // W staged *transposed*: Wt[n][k], same stride
#define XS_ELEMS  (128 * XS_STRIDE)  // 4608 floats  (18,432 B)
#define WT_ELEMS  (D_OUT * WT_STRIDE)// 9216 floats  (36,864 B)

typedef float v2f __attribute__((ext_vector_type(2)));
typedef float v8f __attribute__((ext_vector_type(8)));

// addrspace(3) view of a pointer -> raw LDS byte offset for async-DMA operands
typedef __attribute__((address_space(3))) float lds_f;
__device__ __forceinline__ unsigned lds_off(const float* p) {
    return (unsigned)(unsigned long long)(lds_f*)p;
}

__global__ __launch_bounds__(256, 1)
void adaptive_linear_wmma_f32(const float* __restrict__ x,
                              const int*   __restrict__ indices,
                              const float* __restrict__ weight,
                              const float* __restrict__ bias,
                              float*       __restrict__ out)
{
    // Double-buffered slabs: 2*(18432 + 36864) = 110,592 bytes of LDS
    __shared__ float Xs[2][XS_ELEMS];
    __shared__ float Wt[2][WT_ELEMS];

    const int tile   = blockIdx.y;          // 0..511
    const int mblock = blockIdx.x;          // 0..3 -> 128-row panel
    const int tid    = threadIdx.x;         // 256 threads = 8 waves
    const int lane   = tid & 31;
    const int wave   = tid >> 5;
    const int wm     = wave >> 2;           // 0..1 : 64-row half
    const int wn     = wave & 3;            // 0..3 : 64-col quarter
    const int lhalf  = lane >> 4;
    const int l16    = lane & 15;

    const int ch = indices[tile];
    const unsigned long long xg = (unsigned long long)
        (x + ((size_t)tile * P_PTS + (size_t)mblock * 128) * D_IN);
    const unsigned long long wg = (unsigned long long)
        (weight + (size_t)ch * D_IN * D_OUT);

    v8f acc[4][4];
    #pragma unroll
    for (int mt = 0; mt < 4; ++mt)
        #pragma unroll
        for (int nt = 0; nt < 4; ++nt)
            acc[mt][nt] = (v8f){0.f,0.f,0.f,0.f,0.f,0.f,0.f,0.f};

    // Async-DMA one K-slab into LDS buffer `buf`:
    //   X: 128x32 row-major (b128 transfers, 4/thread)
    //   W: 32x256 transposed into Wt[n][k] (b32 scatter, 32/thread) so B
    //      fragments become contiguous per-lane float2 like A fragments.
    auto stage = [&](int buf, int kb) {
        #pragma unroll
        for (int j = 0; j < 4; ++j) {               // X: 1024 b128 total
            int i   = tid + 256 * j;
            int row = i >> 3;                       // 8 x float4 per row
            int col = (i & 7) << 2;
            unsigned lds = lds_off(&Xs[buf][row * XS_STRIDE + col]);
            unsigned off = (unsigned)((row * D_IN + kb + col) * 4);
            asm volatile("global_load_async_to_lds_b128 %0, %1, %2"
                         :: "v"(lds), "v"(off), "s"(xg) : "memory");
        }
        #pragma unroll
        for (int j = 0; j < 32; ++j) {              // W: 8192 b32 total
            int e = tid + 256 * j;                  // element id 0..8191
            int k = e >> 8;                         // 0..31
            int n = e & 255;                        // consecutive across lanes
            unsigned lds = lds_off(&Wt[buf][n * WT_STRIDE + k]);
            unsigned off = (unsigned)(((kb + k) * D_OUT + n) * 4);
            asm volatile("global_load_async_to_lds_b32 %0, %1, %2"
                         :: "v"(lds), "v"(off), "s"(wg) : "memory");
        }
    };

    // Prologue: fill buffer 0
    stage(0, 0);
    asm volatile("s_wait_asynccnt 0x0" ::: "memory");
    __syncthreads();

    for (int kbi = 0; kbi < D_IN / KB; ++kbi) {
        const int cur = kbi & 1;
        // Kick off DMA for the next slab before computing on this one
        if (kbi + 1 < D_IN / KB)
            stage(cur ^ 1, (kbi + 1) * KB);

        const float* Xc = Xs[cur];
        const float* Wc = Wt[cur];

        #pragma unroll
        for (int k0 = 0; k0 < KB; k0 += 4) {
            // A fragment (16x4 f32): lanes 0-15 K={k0,k0+1}, lanes 16-31 K={k0+2,k0+3}
            v2f a[4];
            #pragma unroll
            for (int mt = 0; mt < 4; ++mt) {
                int row = wm * 64 + mt * 16 + l16;
                a[mt] = *(const v2f*)(&Xc[row * XS_STRIDE + k0 + (lhalf << 1)]);
            }
            // B fragment (4x16 f32): with Wt[n][k] this is the same contiguous
            // per-lane float2 shape as A -> single ds_load_b64 per fragment.
            v2f b[4];
            #pragma unroll
            for (int nt = 0; nt < 4; ++nt) {
                int coln = wn * 64 + nt * 16 + l16;
                b[nt] = *(const v2f*)(&Wc[coln * WT_STRIDE + k0 + (lhalf << 1)]);
            }
            #pragma unroll
            for (int mt = 0; mt < 4; ++mt)
                #pragma unroll
                for (int nt = 0; nt < 4; ++nt)
                    acc[mt][nt] = __builtin_amdgcn_wmma_f32_16x16x4_f32(
                        false, a[mt], false, b[nt],
                        (short)0, acc[mt][nt], false, false);
        }

        // My async writes into `next` done; everyone done reading `cur`.
        asm volatile("s_wait_asynccnt 0x0" ::: "memory");
        __syncthreads();
    }

    // Epilogue: bias add + store. C/D: VGPR r -> M=r (lanes<16) / r+8, N=l16.
    float bv[4];
    #pragma unroll
    for (int nt = 0; nt < 4; ++nt)
        bv[nt] = bias[wn * 64 + nt * 16 + l16];

    float* ob = out + ((size_t)tile * P_PTS + (size_t)mblock * 128) * D_OUT;
    #pragma unroll
    for (int mt = 0; mt < 4; ++mt) {
        #pragma unroll
        for (int nt = 0; nt < 4; ++nt) {
            int col = wn * 64 + nt * 16 + l16;
            #pragma unroll
            for (int r = 0; r < 8; ++r) {
                int row = wm * 64 + mt * 16 + r + (lhalf << 3);
                ob[(size_t)row * D_OUT + col] = acc[mt][nt][r] + bv[nt];
            }
        }
    }
}

extern "C" void kernel_launch(void* const* d_in, const int* in_sizes, int n_in,
                              void* d_out, int out_size, void* d_ws, size_t ws_size,
                              hipStream_t stream)
{
    const float* x       = (const float*)d_in[0];
    const int*   indices = (const int*)d_in[1];
    const float* weight  = (const float*)d_in[2];
    const float* bias    = (const float*)d_in[3];
    float*       out     = (float*)d_out;

    dim3 grid(P_PTS / 128, T_TILES);  // (4, 512)
    dim3 block(256);
    adaptive_linear_wmma_f32<<<grid, block, 0, stream>>>(x, indices, weight, bias, out);
}